// GraphSAGE_52673478918497
// MI455X (gfx1250) — compile-verified
//
#include <hip/hip_runtime.h>

// ---------------------------------------------------------------------------
// GraphSAGE 2-layer forward for MI455X (gfx1250, wave32).
//   layer: out = relu?( mean_agg(x) @ W_l.T + b + x @ W_r.T )
// Scatter phase: L2-resident fp32 atomics (dominant cost, ~1.2 GB L2 traffic).
// Dense phase: V_WMMA_F32_16X16X4_F32, one wave per 16-node tile.
// ---------------------------------------------------------------------------

typedef __attribute__((ext_vector_type(2))) float v2f;
typedef __attribute__((ext_vector_type(4))) float v4f;
typedef __attribute__((ext_vector_type(8))) float v8f;

#define NODES 100000
#define EDGES 1600000
#define INF   32
#define HIDF  64
#define OUTF  8

__device__ __forceinline__ v8f wmma4(v2f a, v2f b, v8f c) {
  // D(16x16,f32) = A(16x4,f32) * B(4x16,f32) + C   -- v_wmma_f32_16x16x4_f32
  return __builtin_amdgcn_wmma_f32_16x16x4_f32(
      /*neg_a=*/false, a, /*neg_b=*/false, b,
      /*c_mod=*/(short)0, c, /*reuse_a=*/false, /*reuse_b=*/false);
}

// ---------------- zero workspace (deg + msg1 + msg2) -----------------------
__global__ void k_zero(v4f* __restrict__ p, int n4) {
  int i = blockIdx.x * blockDim.x + threadIdx.x;
  if (i < n4) {
    v4f z = {0.f, 0.f, 0.f, 0.f};
    p[i] = z;
  }
}

// ---------------- degree ---------------------------------------------------
__global__ void k_deg(const int* __restrict__ dst, float* __restrict__ deg) {
  int e = blockIdx.x * blockDim.x + threadIdx.x;
  if (e < EDGES) atomicAdd(&deg[dst[e]], 1.0f);
}

// ---------------- edge scatter-add: msg[dst] += feat[src] ------------------
// One lane per (edge, feature). dim = 1<<ld features per node; waves are
// aligned so a wave covers one edge's contiguous row -> coalesced loads,
// atomics land in L2 (node tables fit in 192MB L2).
__global__ void k_scatter(const float* __restrict__ feat,
                          const int* __restrict__ src,
                          const int* __restrict__ dst,
                          float* __restrict__ msg,
                          int ld, int total) {
  int t = blockIdx.x * blockDim.x + threadIdx.x;
  if (t >= total) return;
  int e = t >> ld;
  int f = t & ((1 << ld) - 1);
  int s = src[e];
  int d = dst[e];
  atomicAdd(&msg[(d << ld) + f], feat[(s << ld) + f]);
}

// ---------------- layer 1: h = relu(agg@W1l.T + b1 + x@W1r.T) --------------
// K=32, Nout=64. One wave -> 16 nodes, four 16x16 N-tiles, 16 WMMA each.
__global__ void __launch_bounds__(256)
k_layer1(const float* __restrict__ x, const float* __restrict__ msg1,
         const float* __restrict__ deg, const float* __restrict__ Wl,
         const float* __restrict__ b, const float* __restrict__ Wr,
         float* __restrict__ h) {
  __shared__ __align__(16) float sWl[HIDF * INF];
  __shared__ __align__(16) float sWr[HIDF * INF];
  __shared__ __align__(16) float sB[HIDF];
  for (int i = threadIdx.x; i < HIDF * INF; i += 256) {
    sWl[i] = Wl[i];
    sWr[i] = Wr[i];
  }
  if (threadIdx.x < HIDF) sB[threadIdx.x] = b[threadIdx.x];
  __syncthreads();

  const int lane  = threadIdx.x & 31;
  const int wave  = threadIdx.x >> 5;
  const int tile  = blockIdx.x * 8 + wave;
  const int node0 = tile * 16;
  if (node0 >= NODES) return;  // wave-uniform; EXEC stays all-1s for WMMA

  const int m  = lane & 15;   // M row (A load) / N col (B load, C store)
  const int kh = lane >> 4;   // K half-pair select per ISA A/B layouts

  int row = node0 + m;
  if (row >= NODES) row = NODES - 1;  // 100000 % 16 == 0: never taken
  const float invd = 1.0f / fmaxf(deg[row], 1.0f);

  const v2f* aggp = (const v2f*)(msg1 + row * INF);
  const v2f* xp   = (const v2f*)(x + row * INF);
  v2f Aa[8], Ax[8];
#pragma unroll
  for (int j = 0; j < 8; ++j) {          // k0 = 4*j ; lane holds {k0+2kh, k0+2kh+1}
    Aa[j] = aggp[2 * j + kh] * invd;     // fold mean into A fragment
    Ax[j] = xp[2 * j + kh];
  }

#pragma unroll
  for (int nt = 0; nt < 4; ++nt) {
    const int n0 = nt * 16;
    const v2f* wl = (const v2f*)(sWl + (n0 + m) * INF);
    const v2f* wr = (const v2f*)(sWr + (n0 + m) * INF);
    v8f acc = {0.f, 0.f, 0.f, 0.f, 0.f, 0.f, 0.f, 0.f};
#pragma unroll
    for (int j = 0; j < 8; ++j) acc = wmma4(Aa[j], wl[2 * j + kh], acc);
#pragma unroll
    for (int j = 0; j < 8; ++j) acc = wmma4(Ax[j], wr[2 * j + kh], acc);

    const float bias = sB[n0 + m];
#pragma unroll
    for (int v = 0; v < 8; ++v) {        // C layout: M = v + 8*kh, N = m
      int orow = node0 + v + 8 * kh;
      float val = fmaxf(acc[v] + bias, 0.0f);
      if (orow < NODES) h[orow * HIDF + n0 + m] = val;
    }
  }
}

// ---------------- layer 2: out = agg2@W2l.T + b2 + h@W2r.T -----------------
// K=64, Nout=8 (pad B cols 8..15 with zeros, mask the store).
__global__ void __launch_bounds__(256)
k_layer2(const float* __restrict__ hin, const float* __restrict__ msg2,
         const float* __restrict__ deg, const float* __restrict__ Wl,
         const float* __restrict__ b, const float* __restrict__ Wr,
         float* __restrict__ out) {
  __shared__ __align__(16) float sWl[OUTF * HIDF];
  __shared__ __align__(16) float sWr[OUTF * HIDF];
  __shared__ __align__(16) float sB[OUTF];
  for (int i = threadIdx.x; i < OUTF * HIDF; i += 256) {
    sWl[i] = Wl[i];
    sWr[i] = Wr[i];
  }
  if (threadIdx.x < OUTF) sB[threadIdx.x] = b[threadIdx.x];
  __syncthreads();

  const int lane  = threadIdx.x & 31;
  const int wave  = threadIdx.x >> 5;
  const int tile  = blockIdx.x * 8 + wave;
  const int node0 = tile * 16;
  if (node0 >= NODES) return;

  const int m  = lane & 15;
  const int kh = lane >> 4;
  const bool ncol = (m < OUTF);
  const int nc = ncol ? m : 0;           // clamp; zero the fragment instead

  int row = node0 + m;
  if (row >= NODES) row = NODES - 1;
  const float invd = 1.0f / fmaxf(deg[row], 1.0f);

  const v2f* aggp = (const v2f*)(msg2 + row * HIDF);
  const v2f* hp   = (const v2f*)(hin + row * HIDF);
  const v2f* wl   = (const v2f*)(sWl + nc * HIDF);
  const v2f* wr   = (const v2f*)(sWr + nc * HIDF);
  const v2f z2 = {0.f, 0.f};

  v8f acc = {0.f, 0.f, 0.f, 0.f, 0.f, 0.f, 0.f, 0.f};
#pragma unroll
  for (int j = 0; j < 16; ++j) {         // K = 64 -> 16 steps of 4
    v2f bl = ncol ? wl[2 * j + kh] : z2; // lane-level select, EXEC untouched
    acc = wmma4(aggp[2 * j + kh] * invd, bl, acc);
  }
#pragma unroll
  for (int j = 0; j < 16; ++j) {
    v2f br = ncol ? wr[2 * j + kh] : z2;
    acc = wmma4(hp[2 * j + kh], br, acc);
  }

  const float bias = ncol ? sB[nc] : 0.0f;
#pragma unroll
  for (int v = 0; v < 8; ++v) {          // C: M = v + 8*kh, N = m
    int orow = node0 + v + 8 * kh;
    if (ncol && orow < NODES) out[orow * OUTF + m] = acc[v] + bias;
  }
}

// ---------------------------------------------------------------------------
extern "C" void kernel_launch(void* const* d_in, const int* in_sizes, int n_in,
                              void* d_out, int out_size, void* d_ws, size_t ws_size,
                              hipStream_t stream) {
  (void)in_sizes; (void)n_in; (void)out_size; (void)ws_size;

  const float* x   = (const float*)d_in[0];
  const int*   ei  = (const int*)d_in[1];     // [2, EDGES] flat
  const int*   src = ei;
  const int*   dst = ei + EDGES;
  const float* W1l = (const float*)d_in[2];
  const float* b1  = (const float*)d_in[3];
  const float* W1r = (const float*)d_in[4];
  const float* W2l = (const float*)d_in[5];
  const float* b2  = (const float*)d_in[6];
  const float* W2r = (const float*)d_in[7];
  float* out = (float*)d_out;

  float* ws   = (float*)d_ws;
  float* deg  = ws;                 // [NODES]
  float* msg1 = ws + NODES;         // [NODES*32]
  float* msg2 = ws + NODES * 33;    // [NODES*64]
  float* hbuf = ws + NODES * 97;    // [NODES*64] (fully overwritten, no zeroing)

  // zero deg|msg1|msg2 = NODES*97 floats (multiple of 4)
  const int n4 = (NODES * 97) / 4;
  k_zero<<<(n4 + 255) / 256, 256, 0, stream>>>((v4f*)ws, n4);

  k_deg<<<(EDGES + 255) / 256, 256, 0, stream>>>(dst, deg);

  const int tot1 = EDGES * INF;     // 51.2M lanes
  k_scatter<<<(tot1 + 255) / 256, 256, 0, stream>>>(x, src, dst, msg1, 5, tot1);

  const int tiles  = NODES / 16;    // 6250
  const int blocks = (tiles + 7) / 8;
  k_layer1<<<blocks, 256, 0, stream>>>(x, msg1, deg, W1l, b1, W1r, hbuf);

  const int tot2 = EDGES * HIDF;    // 102.4M lanes
  k_scatter<<<(tot2 + 255) / 256, 256, 0, stream>>>(hbuf, src, dst, msg2, 6, tot2);

  k_layer2<<<blocks, 256, 0, stream>>>(hbuf, msg2, deg, W2l, b2, W2r, out);
}